// Router_42271068127817
// MI455X (gfx1250) — compile-verified
//
#include <hip/hip_runtime.h>
#include <hip/hip_bf16.h>

typedef __attribute__((ext_vector_type(2))) float v2f;
typedef __attribute__((ext_vector_type(8))) float v8f;

#define B_SZ   128
#define I_CAPS 1152
#define J_CAPS 10
#define N_DIM  8
#define M_DIM  16
#define ITER   3
#define EPS_F  1e-8f

// -----------------------------------------------------------------------------
// Kernel 1: u_hat[b,i,j,m] = sum_n w[i,j,n,m] * x[b,i,n]
// One wave per (i, btile); the 16x8 A-tile (x) is loaded once and reused for
// all 10 j-tiles: 20 x V_WMMA_F32_16X16X4_F32 per wave (K=8 as 2 k-steps).
// All store addresses are base + compile-time-constant offsets (fold into the
// 24-bit IOFFSET field: r*737280 B + j*64 B, max ~5.2 MB < +-8 MB).
// -----------------------------------------------------------------------------
__global__ __launch_bounds__(256) void uhat_wmma_kernel(
    const float* __restrict__ x,    // [B, I, N]
    const float* __restrict__ w,    // [I, J, N, M]
    float* __restrict__ uhat)       // [B, I, J, M]
{
    const int wave = threadIdx.x >> 5;
    const int lane = threadIdx.x & 31;

    // tile = i * 8 + btile ; total tiles = 1152*8 = 9216 (1152 blocks x 8 waves)
    const int tile  = blockIdx.x * 8 + wave;
    const int btile = tile & 7;      // B/16 = 8 tiles
    const int i     = tile >> 3;

    const int rc    = lane & 15;     // row (A) / col (B,D) within tile
    const int khalf = lane >> 4;     // 0: k-pair {0,1}, 1: k-pair {2,3}

    // ---- A-tile: x[btile*16 + rc, i, khalf*2 + {0,1}] (+4 for 2nd k-step) ----
    const float* xa =
        x + ((size_t)(btile * 16 + rc) * I_CAPS + i) * N_DIM + khalf * 2;
    const v2f a0 = *(const v2f*)(xa);      // 8B-aligned (row base is 32B-aligned)
    const v2f a1 = *(const v2f*)(xa + 4);

    // ---- B base: w[i, 0, khalf*2, rc]; per-j stride = N*M = 128 floats ----
    const float* wb =
        w + (size_t)i * J_CAPS * N_DIM * M_DIM + (khalf * 2) * M_DIM + rc;

    // ---- D base: uhat[btile*16 + khalf*8, i, 0, rc] ----
    float* ob =
        uhat + (((size_t)(btile * 16 + khalf * 8) * I_CAPS + i) * J_CAPS) * M_DIM
             + rc;

    #pragma unroll
    for (int j = 0; j < J_CAPS; ++j) {
        const float* wj = wb + j * (N_DIM * M_DIM);
        v2f b0, b1;
        b0.x = wj[0 * M_DIM];                // k = khalf*2
        b0.y = wj[1 * M_DIM];                // k = khalf*2 + 1
        b1.x = wj[4 * M_DIM];                // k = 4 + khalf*2
        b1.y = wj[5 * M_DIM];                // k = 4 + khalf*2 + 1

        v8f acc = {};
        acc = __builtin_amdgcn_wmma_f32_16x16x4_f32(
            false, a0, false, b0, (short)0, acc, false, false);
        acc = __builtin_amdgcn_wmma_f32_16x16x4_f32(
            false, a1, false, b1, (short)0, acc, false, false);

        // D layout: VGPR r -> row (b) = khalf*8 + r, col = rc
        #pragma unroll
        for (int r = 0; r < 8; ++r) {
            ob[(size_t)r * (I_CAPS * J_CAPS * M_DIM) + j * M_DIM] = acc[r];
        }
    }
}

// -----------------------------------------------------------------------------
// Kernel 2: dynamic routing, one 256-thread block per batch element.
// logits[1152][10] live in LDS; 3 iterations; last iteration skips the dead
// a-pass/logit update and writes v.
// -----------------------------------------------------------------------------
__global__ __launch_bounds__(256) void routing_kernel(
    const float* __restrict__ uhat,  // [B, I, J, M]
    float* __restrict__ out)         // [B, J, M]
{
    __shared__ float logits[I_CAPS * J_CAPS];   // 46080 B
    __shared__ float red[J_CAPS * 256];         // 10240 B
    __shared__ float sj[J_CAPS * M_DIM];
    __shared__ float vj[J_CAPS * M_DIM];
    __shared__ float maxj[J_CAPS];
    __shared__ float invden[J_CAPS];

    const int tid = threadIdx.x;
    const int b   = blockIdx.x;
    const float* ub = uhat + (size_t)b * I_CAPS * J_CAPS * M_DIM;

    for (int p = tid; p < I_CAPS * J_CAPS; p += 256) logits[p] = 0.0f;
    __syncthreads();

    for (int iter = 0; iter < ITER; ++iter) {
        // ---- softmax statistics over i, all 10 j's reduced in parallel ----
        if (iter > 0) {
            float mx[J_CAPS];
            #pragma unroll
            for (int j = 0; j < J_CAPS; ++j) mx[j] = -3.4e38f;
            for (int i = tid; i < I_CAPS; i += 256) {
                #pragma unroll
                for (int j = 0; j < J_CAPS; ++j)
                    mx[j] = fmaxf(mx[j], logits[i * J_CAPS + j]);
            }
            #pragma unroll
            for (int j = 0; j < J_CAPS; ++j) red[j * 256 + tid] = mx[j];
            __syncthreads();
            for (int s = 128; s > 0; s >>= 1) {
                if (tid < s) {
                    #pragma unroll
                    for (int j = 0; j < J_CAPS; ++j)
                        red[j * 256 + tid] =
                            fmaxf(red[j * 256 + tid], red[j * 256 + tid + s]);
                }
                __syncthreads();
            }
            float mjv[J_CAPS];
            #pragma unroll
            for (int j = 0; j < J_CAPS; ++j) mjv[j] = red[j * 256];
            __syncthreads();

            float se[J_CAPS];
            #pragma unroll
            for (int j = 0; j < J_CAPS; ++j) se[j] = 0.0f;
            for (int i = tid; i < I_CAPS; i += 256) {
                #pragma unroll
                for (int j = 0; j < J_CAPS; ++j)
                    se[j] += __expf(logits[i * J_CAPS + j] - mjv[j]);
            }
            #pragma unroll
            for (int j = 0; j < J_CAPS; ++j) red[j * 256 + tid] = se[j];
            __syncthreads();
            for (int s = 128; s > 0; s >>= 1) {
                if (tid < s) {
                    #pragma unroll
                    for (int j = 0; j < J_CAPS; ++j)
                        red[j * 256 + tid] += red[j * 256 + tid + s];
                }
                __syncthreads();
            }
            if (tid < J_CAPS) {
                maxj[tid]   = mjv[tid];
                invden[tid] = 1.0f / red[tid * 256];
            }
            __syncthreads();
        }

        // ---- s-pass: thread (j,m) accumulates over all i (coalesced rows) ----
        if (tid < J_CAPS * M_DIM) {
            const int j = tid >> 4, m = tid & 15;
            const float mj  = (iter > 0) ? maxj[j]   : 0.0f;
            const float idn = (iter > 0) ? invden[j] : (1.0f / I_CAPS);
            float acc = 0.0f;
            for (int i = 0; i < I_CAPS; ++i) {
                const float c = (iter > 0)
                    ? __expf(logits[i * J_CAPS + j] - mj) * idn
                    : idn;
                acc += c * ub[((size_t)i * J_CAPS + j) * M_DIM + m];
            }
            sj[tid] = acc;
        }
        __syncthreads();

        // ---- squash per output capsule j ----
        if (tid < J_CAPS) {
            float s2 = 0.0f;
            #pragma unroll
            for (int m = 0; m < M_DIM; ++m) {
                const float t = sj[tid * M_DIM + m];
                s2 += t * t;
            }
            const float scale = (s2 / (1.0f + s2)) * rsqrtf(s2 + EPS_F);
            #pragma unroll
            for (int m = 0; m < M_DIM; ++m)
                vj[tid * M_DIM + m] = scale * sj[tid * M_DIM + m];
        }
        __syncthreads();

        if (iter == ITER - 1) {
            if (tid < J_CAPS * M_DIM)
                out[(size_t)b * J_CAPS * M_DIM + tid] = vj[tid];
        } else {
            // ---- a-pass: logits[i,j] += dot(v[j,:], u_hat[b,i,j,:]) ----
            for (int p = tid; p < I_CAPS * J_CAPS; p += 256) {
                const int i = p / J_CAPS, j = p % J_CAPS;
                const float4* up =
                    (const float4*)(ub + ((size_t)i * J_CAPS + j) * M_DIM);
                float a = 0.0f;
                #pragma unroll
                for (int q = 0; q < 4; ++q) {
                    const float4 u4 = up[q];
                    a += vj[j * 16 + q * 4 + 0] * u4.x
                       + vj[j * 16 + q * 4 + 1] * u4.y
                       + vj[j * 16 + q * 4 + 2] * u4.z
                       + vj[j * 16 + q * 4 + 3] * u4.w;
                }
                logits[p] += a;
            }
            __syncthreads();
        }
    }
}

extern "C" void kernel_launch(void* const* d_in, const int* in_sizes, int n_in,
                              void* d_out, int out_size, void* d_ws, size_t ws_size,
                              hipStream_t stream) {
    // setup_inputs order: nums_caps (int), out_caps (int), x (f32), w (f32)
    const float* x = (const float*)d_in[2];
    const float* w = (const float*)d_in[3];
    float* out  = (float*)d_out;
    float* uhat = (float*)d_ws;  // needs 128*1152*10*16*4 = 94,371,840 bytes

    // 9216 waves (one per (i, btile)), 8 waves per 256-thread block
    uhat_wmma_kernel<<<dim3(I_CAPS), dim3(256), 0, stream>>>(x, w, uhat);
    routing_kernel<<<dim3(B_SZ), dim3(256), 0, stream>>>(uhat, out);
}